// GraphWaveletTransform_5325759447103
// MI455X (gfx1250) — compile-verified
//
#include <hip/hip_runtime.h>
#include <hip/hip_bf16.h>

// ---------------------------------------------------------------------------
// Graph wavelet scattering transform, MI455X (gfx1250, wave32).
//
//  * Diffusion is linear & column-independent -> diffusing U=[D1|D2|D4|D8]
//    equals diffusing each D_j separately; D8's diffusion is never needed.
//    => 4 diffusions x 8 steps on [N,32] blocks (32 propagate steps total).
//  * Pull-based CSR by destination (built once per call, edge order made
//    deterministic by per-row sort on edge id) -> zero float atomics in the
//    hot loop. One wave per node, lane = feature column, 128B coalesced
//    gathers, 4x unrolled for MLP; working set (~75MB) lives in the 192MB L2.
//  * Segment-mean pooling with V_WMMA_F32_16X16X4_F32: A = ones(16x4),
//    B = 4 nodes x 16 cols, f32 accumulation (exact). Two independent
//    accumulators (8 rows/iter) so consecutive WMMAs have no RAW dependency;
//    branchless masked tail keeps EXEC all-ones at every WMMA.
// ---------------------------------------------------------------------------

typedef __attribute__((ext_vector_type(2))) float v2f;
typedef __attribute__((ext_vector_type(8))) float v8f;

#define N_NODES_   50000
#define D_FEAT_    32
#define N_GRAPHS_  64
#define N_COLS_OUT 320

// ---------------- CSR construction --------------------------------------

__global__ void hist_kernel(const int* __restrict__ dst, int* __restrict__ deg, int E) {
    int e = blockIdx.x * blockDim.x + threadIdx.x;
    if (e < E) atomicAdd(&deg[dst[e]], 1);
}

// Single-block chunked scan: exclusive prefix of deg -> rofs, copy -> cursor.
__global__ void scan_kernel(const int* __restrict__ deg, int* __restrict__ rofs,
                            int* __restrict__ cursor, int n) {
    __shared__ int carry;
    __shared__ int tmp[1024];
    if (threadIdx.x == 0) carry = 0;
    __syncthreads();
    for (int base = 0; base < n; base += 1024) {
        int i = base + (int)threadIdx.x;
        int v = (i < n) ? deg[i] : 0;
        tmp[threadIdx.x] = v;
        __syncthreads();
        for (int off = 1; off < 1024; off <<= 1) {
            int t = (threadIdx.x >= (unsigned)off) ? tmp[threadIdx.x - off] : 0;
            __syncthreads();
            tmp[threadIdx.x] += t;
            __syncthreads();
        }
        int incl = tmp[threadIdx.x];
        int excl = incl - v;
        if (i < n) { int o = carry + excl; rofs[i] = o; cursor[i] = o; }
        __syncthreads();
        if (threadIdx.x == 1023) carry += incl;
        __syncthreads();
    }
    if (threadIdx.x == 0) rofs[n] = carry;
}

__global__ void fill_eid_kernel(const int* __restrict__ dst, int* __restrict__ cursor,
                                int* __restrict__ ceid, int E) {
    int e = blockIdx.x * blockDim.x + threadIdx.x;
    if (e < E) {
        int pos = atomicAdd(&cursor[dst[e]], 1);
        ceid[pos] = e;
    }
}

// Deterministic row order: insertion-sort edge ids within each row (avg deg ~16).
__global__ void sort_rows_kernel(int* __restrict__ ceid, const int* __restrict__ rofs, int n) {
    int r = blockIdx.x * blockDim.x + threadIdx.x;
    if (r >= n) return;
    int beg = rofs[r], end = rofs[r + 1];
    for (int i = beg + 1; i < end; ++i) {
        int v = ceid[i];
        int j = i - 1;
        while (j >= beg && ceid[j] > v) { ceid[j + 1] = ceid[j]; --j; }
        ceid[j + 1] = v;
    }
}

__global__ void csr_gather_kernel(const int* __restrict__ ceid, const int* __restrict__ src,
                                  const float* __restrict__ w, int* __restrict__ csrc,
                                  float* __restrict__ cw, int E) {
    int k = blockIdx.x * blockDim.x + threadIdx.x;
    if (k < E) {
        int e = ceid[k];
        csrc[k] = src[e];
        cw[k]   = w[e];
    }
}

// batch is sorted; gofs[g] = lower_bound(batch, g)
__global__ void graph_ofs_kernel(const int* __restrict__ batch, int* __restrict__ gofs,
                                 int n_nodes, int n_graphs) {
    int g = blockIdx.x * blockDim.x + threadIdx.x;
    if (g > n_graphs) return;
    int lo = 0, hi = n_nodes;
    while (lo < hi) { int mid = (lo + hi) >> 1; if (batch[mid] < g) lo = mid + 1; else hi = mid; }
    gofs[g] = lo;
}

// ---------------- Propagate: out[n,:] = sum_{e: dst=n} w_e * in[src_e,:] ---

__global__ __launch_bounds__(256) void prop_kernel(const float* __restrict__ xin,
                                                   float* __restrict__ xout,
                                                   const int* __restrict__ rofs,
                                                   const int* __restrict__ csrc,
                                                   const float* __restrict__ cw,
                                                   int n_nodes) {
    const int lane = threadIdx.x & 31;
    int wave = (int)(blockIdx.x * (blockDim.x >> 5)) + (int)(threadIdx.x >> 5);
    int node = __builtin_amdgcn_readfirstlane(wave);
    if (node >= n_nodes) return;
    int beg = __builtin_amdgcn_readfirstlane(rofs[node]);
    int end = __builtin_amdgcn_readfirstlane(rofs[node + 1]);
    float acc = 0.0f;
    int k = beg;
    // 4x unrolled: four independent 128B gathers in flight per wave.
    for (; k + 4 <= end; k += 4) {
        int s0 = __builtin_amdgcn_readfirstlane(csrc[k]);
        int s1 = __builtin_amdgcn_readfirstlane(csrc[k + 1]);
        int s2 = __builtin_amdgcn_readfirstlane(csrc[k + 2]);
        int s3 = __builtin_amdgcn_readfirstlane(csrc[k + 3]);
        if (k + 4 < end) {
            int sn = __builtin_amdgcn_readfirstlane(csrc[k + 4]);
            __builtin_prefetch(&xin[(size_t)sn * D_FEAT_ + lane], 0, 1); // global_prefetch_b8
        }
        float w0 = cw[k], w1 = cw[k + 1], w2 = cw[k + 2], w3 = cw[k + 3];
        float x0 = xin[(size_t)s0 * D_FEAT_ + lane];
        float x1 = xin[(size_t)s1 * D_FEAT_ + lane];
        float x2 = xin[(size_t)s2 * D_FEAT_ + lane];
        float x3 = xin[(size_t)s3 * D_FEAT_ + lane];
        acc = fmaf(x0, w0, acc);
        acc = fmaf(x1, w1, acc);
        acc = fmaf(x2, w2, acc);
        acc = fmaf(x3, w3, acc);
    }
    for (; k < end; ++k) {
        int s = __builtin_amdgcn_readfirstlane(csrc[k]);
        float wk = cw[k];
        acc = fmaf(xin[(size_t)s * D_FEAT_ + lane], wk, acc);
    }
    xout[(size_t)node * D_FEAT_ + lane] = acc;
}

// ---------------- WMMA segment-mean pooling -------------------------------
// feats block f(n,c) = HASB ? |a[n,c]-b[n,c]| : a[n,c]  for c in [0,32)
// out[g, col_out + c] = mean over nodes of graph g of f(n,c)
// One wave per (graph, 16-col half): D = ones(16x4) x B(4x16) + C.
template <bool HASB>
__device__ __forceinline__ float feat_val(const float* __restrict__ a,
                                          const float* __restrict__ b,
                                          int row, int col) {
    float t = a[(size_t)row * D_FEAT_ + col];
    if (HASB) t = fabsf(t - b[(size_t)row * D_FEAT_ + col]);
    return t;
}

template <bool HASB>
__global__ __launch_bounds__(256) void pool_wmma_kernel(const float* __restrict__ a,
                                                        const float* __restrict__ b,
                                                        float* __restrict__ out,
                                                        const int* __restrict__ gofs,
                                                        int col_out, int n_graphs) {
    const int lane = threadIdx.x & 31;
    int wave = (int)(blockIdx.x * (blockDim.x >> 5)) + (int)(threadIdx.x >> 5);
    int task = __builtin_amdgcn_readfirstlane(wave);
    if (task >= n_graphs * 2) return;
    int g = task >> 1;
    int h = task & 1; // which 16-column half of this 32-col block
    int gs = __builtin_amdgcn_readfirstlane(gofs[g]);
    int ge = __builtin_amdgcn_readfirstlane(gofs[g + 1]);

    int ncol = lane & 15;
    int kofs = (lane >> 4) << 1;          // A/B VGPR layout: lanes16-31 hold K=2,3
    int coff = h * 16 + ncol;             // column within the 32-wide block

    v2f A; A.x = 1.0f; A.y = 1.0f;        // ones(16x4)
    v8f c0 = {};
    v8f c1 = {};

    int nb = gs;
    // Main: 8 rows/iter, two independent accumulators -> independent WMMAs.
    for (; nb + 8 <= ge; nb += 8) {
        v2f B0, B1;
        B0.x = feat_val<HASB>(a, b, nb + kofs,     coff);
        B0.y = feat_val<HASB>(a, b, nb + kofs + 1, coff);
        B1.x = feat_val<HASB>(a, b, nb + 4 + kofs,     coff);
        B1.y = feat_val<HASB>(a, b, nb + 4 + kofs + 1, coff);
        c0 = __builtin_amdgcn_wmma_f32_16x16x4_f32(false, A, false, B0,
                                                   (short)0, c0, false, false);
        c1 = __builtin_amdgcn_wmma_f32_16x16x4_f32(false, A, false, B1,
                                                   (short)0, c1, false, false);
    }
    // Mid: 4 rows, unconditional.
    for (; nb + 4 <= ge; nb += 4) {
        v2f B;
        B.x = feat_val<HASB>(a, b, nb + kofs,     coff);
        B.y = feat_val<HASB>(a, b, nb + kofs + 1, coff);
        c0 = __builtin_amdgcn_wmma_f32_16x16x4_f32(false, A, false, B,
                                                   (short)0, c0, false, false);
    }
    // Tail: <=3 rows, branchless masked (clamped index + select), EXEC intact.
    if (nb < ge) {                        // wave-uniform condition
        int last = ge - 1;
        int r0 = nb + kofs, r1 = r0 + 1;
        int i0 = r0 < ge ? r0 : last;
        int i1 = r1 < ge ? r1 : last;
        float t0 = feat_val<HASB>(a, b, i0, coff);
        float t1 = feat_val<HASB>(a, b, i1, coff);
        v2f B;
        B.x = r0 < ge ? t0 : 0.0f;
        B.y = r1 < ge ? t1 : 0.0f;
        c0 = __builtin_amdgcn_wmma_f32_16x16x4_f32(false, A, false, B,
                                                   (short)0, c0, false, false);
    }

    int cnt = ge - gs;
    float inv = 1.0f / (float)(cnt > 0 ? cnt : 1);
    float total = c0[0] + c1[0];          // row M=0 lives in VGPR0, lanes 0..15
    if (lane < 16) {
        out[(size_t)g * N_COLS_OUT + col_out + h * 16 + ncol] = total * inv;
    }
}

// ---------------- Host orchestration --------------------------------------

extern "C" void kernel_launch(void* const* d_in, const int* in_sizes, int n_in,
                              void* d_out, int out_size, void* d_ws, size_t ws_size,
                              hipStream_t stream) {
    const float* X     = (const float*)d_in[0];
    const int*   eidx  = (const int*)d_in[1];
    const float* ew    = (const float*)d_in[2];
    const int*   batch = (const int*)d_in[3];
    float* out = (float*)d_out;

    const int E  = in_sizes[1] / 2;
    const int Nn = in_sizes[0] / D_FEAT_;
    const int* esrc = eidx;
    const int* edst = eidx + E;

    // workspace carve-up
    char* p = (char*)d_ws;
    auto alloc = [&](size_t bytes) -> void* {
        void* r = (void*)p;
        p += (bytes + 255) & ~(size_t)255;
        return r;
    };
    int*   gofs   = (int*)alloc((N_GRAPHS_ + 1) * sizeof(int));
    int*   deg    = (int*)alloc((size_t)Nn * sizeof(int));
    int*   cursor = (int*)alloc((size_t)Nn * sizeof(int));
    int*   rofs   = (int*)alloc(((size_t)Nn + 1) * sizeof(int));
    int*   ceid   = (int*)alloc((size_t)E * sizeof(int));
    int*   csrc   = (int*)alloc((size_t)E * sizeof(int));
    float* cw     = (float*)alloc((size_t)E * sizeof(float));
    float* B[10];
    for (int i = 0; i < 10; ++i) B[i] = (float*)alloc((size_t)Nn * D_FEAT_ * sizeof(float));
    // B0=D1 B1=D2 B2=D4 B3=D8 B4=T0 B5=T1 B6=S0 B7=S1 B8=S2 B9=S3

    const int TB = 256;
    int eBlocks = (E + TB - 1) / TB;
    int nBlocks = (Nn + TB - 1) / TB;
    int propBlocks = (Nn + 7) / 8;           // 8 waves/block, 1 wave per node
    int poolBlocks = (N_GRAPHS_ * 2 * 32 + TB - 1) / TB;

    // ---- build pull-CSR (deterministic) ----
    hipMemsetAsync(deg, 0, (size_t)Nn * sizeof(int), stream);
    hist_kernel<<<eBlocks, TB, 0, stream>>>(edst, deg, E);
    scan_kernel<<<1, 1024, 0, stream>>>(deg, rofs, cursor, Nn);
    fill_eid_kernel<<<eBlocks, TB, 0, stream>>>(edst, cursor, ceid, E);
    sort_rows_kernel<<<nBlocks, TB, 0, stream>>>(ceid, rofs, Nn);
    csr_gather_kernel<<<eBlocks, TB, 0, stream>>>(ceid, esrc, ew, csrc, cw, E);
    graph_ofs_kernel<<<1, 128, 0, stream>>>(batch, gofs, Nn, N_GRAPHS_);

    auto prop = [&](const float* in, float* o) {
        prop_kernel<<<propBlocks, TB, 0, stream>>>(in, o, rofs, csrc, cw, Nn);
    };
    auto poolA = [&](const float* a, int col) {
        pool_wmma_kernel<false><<<poolBlocks, TB, 0, stream>>>(a, nullptr, out, gofs,
                                                               col, N_GRAPHS_);
    };
    auto poolAB = [&](const float* a, const float* b, int col) {
        pool_wmma_kernel<true><<<poolBlocks, TB, 0, stream>>>(a, b, out, gofs,
                                                              col, N_GRAPHS_);
    };

    // ---- phase A: diffuse X, collect D1,D2,D4,D8 ----
    prop(X, B[0]);  prop(B[0], B[1]); prop(B[1], B[4]); prop(B[4], B[2]);
    prop(B[2], B[4]); prop(B[4], B[5]); prop(B[5], B[4]); prop(B[4], B[3]);

    // F0 = D8 (cols 0..31); F1 = |D1-D2|,|D2-D4|,|D4-D8| (cols 32..127)
    poolA(B[3], 0);
    poolAB(B[0], B[1], 32);
    poolAB(B[1], B[2], 64);
    poolAB(B[2], B[3], 96);

    // ---- phase C, j=0: diffuse D1, need scales S0..S3 ----
    prop(B[0], B[6]); prop(B[6], B[7]); prop(B[7], B[4]); prop(B[4], B[8]);
    prop(B[8], B[4]); prop(B[4], B[5]); prop(B[5], B[4]); prop(B[4], B[9]);
    poolAB(B[7], B[6], 128);   // |S1-S0|
    poolAB(B[8], B[7], 160);   // |S2-S1|
    poolAB(B[9], B[8], 192);   // |S3-S2|

    // ---- j=1: diffuse D2, need S1,S2,S3 ----
    prop(B[1], B[4]); prop(B[4], B[7]); prop(B[7], B[4]); prop(B[4], B[8]);
    prop(B[8], B[4]); prop(B[4], B[5]); prop(B[5], B[4]); prop(B[4], B[9]);
    poolAB(B[8], B[7], 224);   // |S2-S1|
    poolAB(B[9], B[8], 256);   // |S3-S2|

    // ---- j=2: diffuse D4, need S2,S3 ----
    prop(B[2], B[4]); prop(B[4], B[5]); prop(B[5], B[4]); prop(B[4], B[8]);
    prop(B[8], B[4]); prop(B[4], B[5]); prop(B[5], B[4]); prop(B[4], B[9]);
    poolAB(B[9], B[8], 288);   // |S3-S2|
}